// ChamferLoss_28054726377999
// MI455X (gfx1250) — compile-verified
//
#include <hip/hip_runtime.h>

typedef __attribute__((ext_vector_type(2))) float v2f;
typedef __attribute__((ext_vector_type(8))) float v8f;

#define NPTS 2048
#define NT   128   // number of 16-wide tiles along each cloud

__global__ void chamfer_zero_kernel(float* out) { out[0] = 0.0f; }

__global__ __launch_bounds__(256) void chamfer_wmma_kernel(
    const float* __restrict__ preds,    // (8, 3, 2048, 8) f32
    const float* __restrict__ gts,      // (8, 3, 2048)    f32
    const long long* __restrict__ idx,  // (64, 2)         i64
    float* __restrict__ out)            // scalar
{
  // Staged clouds: xyz + (sum of squares) in .w
  __shared__ float4 sX[NPTS];   // pred-derived cloud
  __shared__ float4 sY[NPTS];   // gt-derived cloud

  const int b    = blockIdx.y;   // batch pair 0..63
  const int dir  = blockIdx.x;   // 0: rows from X (loss_2), 1: rows from Y (loss_1)
  const int half = blockIdx.z;   // row-tile half 0..1

  const long long i0 = idx[2 * b + 0];
  const long long i1 = idx[2 * b + 1];

  const int tid = threadIdx.x;

  // Stage both clouds into LDS (gather from global).
  for (int i = tid; i < NPTS; i += 256) {
    float x0 = preds[(((i0 * 3 + 0) * 2048) + i) * 8 + i1];
    float x1 = preds[(((i0 * 3 + 1) * 2048) + i) * 8 + i1];
    float x2 = preds[(((i0 * 3 + 2) * 2048) + i) * 8 + i1];
    sX[i] = make_float4(x0, x1, x2, x0 * x0 + x1 * x1 + x2 * x2);

    float y0 = gts[(i0 * 3 + 0) * 2048 + i];
    float y1 = gts[(i0 * 3 + 1) * 2048 + i];
    float y2 = gts[(i0 * 3 + 2) * 2048 + i];
    sY[i] = make_float4(y0, y1, y2, y0 * y0 + y1 * y1 + y2 * y2);
  }
  __syncthreads();

  const float4* __restrict__ A = dir ? sY : sX;  // rows (take min over B per row)
  const float4* __restrict__ B = dir ? sX : sY;  // columns

  const int wave = tid >> 5;   // 0..7 (wave32)
  const int lane = tid & 31;
  const int m    = lane & 15;
  const int hi   = lane >> 4;  // which K/row half this lane covers

  float wave_sum = 0.0f;

  for (int rt = half * 64 + wave; rt < half * 64 + 64; rt += 8) {
    // --- A fragment: 16x4 f32, lane holds row M=m, K = 2*hi .. 2*hi+1; K=3 is zero
    float4 pa = A[rt * 16 + m];
    v2f afrag;
    afrag[0] = hi ? pa.z : pa.x;
    afrag[1] = hi ? 0.0f : pa.y;

    // rx for the 8 C-rows this lane owns: row = rt*16 + v + 8*hi
    // (uniform within each 16-lane group; added after the min reduction)
    float rxv[8];
#pragma unroll
    for (int v = 0; v < 8; ++v) rxv[v] = A[rt * 16 + v + 8 * hi].w;

    // minacc tracks min_j (ry_j - 2*zz[i][j]); rx_i is hoisted out of the loop.
    v8f minacc;
#pragma unroll
    for (int v = 0; v < 8; ++v) minacc[v] = 3.0e38f;

#pragma unroll 4
    for (int ct = 0; ct < NT; ++ct) {
      // --- B fragment: 4x16 f32, VGPR0 = K {0|2}, VGPR1 = K {1|3->0}
      float4 pb = B[ct * 16 + m];
      v2f bfrag;
      bfrag[0] = hi ? pb.z : pb.x;
      bfrag[1] = hi ? 0.0f : pb.y;  // K=3 slot: A's K=3 is 0 so B's is don't-care

      v8f c = {};
      v8f zz = __builtin_amdgcn_wmma_f32_16x16x4_f32(
          /*neg_a=*/false, afrag, /*neg_b=*/false, bfrag,
          /*c_mod=*/(short)0, c, /*reuse_a=*/false, /*reuse_b=*/false);

      float ry = pb.w;  // |y_n|^2, n = ct*16 + m (same for both lane halves)
#pragma unroll
      for (int v = 0; v < 8; ++v) {
        float Pp = fmaf(zz[v], -2.0f, ry);   // ry - 2*zz  (rx added later)
        minacc[v] = fminf(minacc[v], Pp);
      }
    }

    // Min-reduce each C element across its 16-lane group (columns of the tile),
    // then add the hoisted rx_i per row.
    float s = 0.0f;
#pragma unroll
    for (int v = 0; v < 8; ++v) {
      float mv = minacc[v];
      mv = fminf(mv, __shfl_xor(mv, 1, 32));
      mv = fminf(mv, __shfl_xor(mv, 2, 32));
      mv = fminf(mv, __shfl_xor(mv, 4, 32));
      mv = fminf(mv, __shfl_xor(mv, 8, 32));
      s += mv + rxv[v];  // lanes 0-15: rows 0-7; lanes 16-31: rows 8-15
    }
    s += __shfl_xor(s, 16, 32);  // all 16 row contributions of this tile
    wave_sum += s;
  }

  if (lane == 0) atomicAdd(out, wave_sum * (1.0f / 64.0f));
}

extern "C" void kernel_launch(void* const* d_in, const int* in_sizes, int n_in,
                              void* d_out, int out_size, void* d_ws, size_t ws_size,
                              hipStream_t stream) {
  const float* preds    = (const float*)d_in[0];
  const float* gts      = (const float*)d_in[1];
  const long long* idx  = (const long long*)d_in[2];
  float* out            = (float*)d_out;

  chamfer_zero_kernel<<<1, 1, 0, stream>>>(out);
  chamfer_wmma_kernel<<<dim3(2, 64, 2), 256, 0, stream>>>(preds, gts, idx, out);
}